// RandomShiftsAug_11630771437601
// MI455X (gfx1250) — compile-verified
//
#include <hip/hip_runtime.h>
#include <stdint.h>

// RandomShiftsAug for MI455X (gfx1250).
// Memory-bound gather: stage each (n, c) 84x84 fp32 plane (28224 B, fully
// contiguous) into LDS with one Tensor Data Mover descriptor, then do the
// reference's bilinear grid-sample math out of LDS. Global traffic is
// read-once + write-once (~231 MB total -> ~10 us at 23.3 TB/s).
// Outputs are written as float4 (84 = 21 * 4 per row, planes 16B-aligned).

typedef unsigned int u32x4 __attribute__((ext_vector_type(4)));
typedef int          i32x8 __attribute__((ext_vector_type(8)));
typedef int          i32x4 __attribute__((ext_vector_type(4)));

#define C_   4
#define H_   84
#define W_   84
#define HW_  (H_ * W_)          // 7056 elements per channel plane
#define PAD_ 4
#define HP_  (H_ + 2 * PAD_)    // 92
#define W4_  (W_ / 4)           // 21 float4 per row
#define NU_  (H_ * W4_)         // 1764 float4 units per plane

__global__ __launch_bounds__(256)
void rsa_tdm_kernel(const float* __restrict__ x,
                    const int*   __restrict__ shift,
                    float*       __restrict__ out) {
    __shared__ float smem[HW_];   // 28224 bytes: one staged channel plane

    const int c = blockIdx.x;
    const int n = blockIdx.y;

    // Per-image shift is uniform across the block -> force into SGPRs.
    int sx = shift[2 * n + 0];
    int sy = shift[2 * n + 1];
    sx = __builtin_amdgcn_readfirstlane(sx);
    sy = __builtin_amdgcn_readfirstlane(sy);

    // ---- TDM: DMA the contiguous (n, c) plane global -> LDS (wave 0 only) ----
    if (threadIdx.x < 32) {
        const uint64_t gaddr = (uint64_t)(uintptr_t)(x + ((size_t)n * C_ + c) * HW_);
        const uint32_t laddr = (uint32_t)(uintptr_t)(&smem[0]);  // LDS byte offset

        // D# group 0 (128b): count=1 | lds_addr | global_addr[56:0] | type=2
        u32x4 g0;
        g0[0] = 1u;                                   // count=1, user mode
        g0[1] = laddr;                                // lds_addr (bytes)
        g0[2] = (uint32_t)gaddr;                      // global_addr[31:0]
        g0[3] = (uint32_t)((gaddr >> 32) & 0x1FFFFFFull) | 0x80000000u; // [56:32] | type=2

        // D# group 1 (256b)
        i32x8 g1;
        g1[0] = (int)(2u << 16);                      // wg_mask=0, data_size=2 (4 bytes)
        g1[1] = (int)((uint32_t)HW_ << 16);           // tensor_dim0[15:0] = 7056
        g1[2] = (int)(1u << 16);                      // tensor_dim0 hi=0 | tensor_dim1 lo = 1
        g1[3] = (int)((uint32_t)HW_ << 16);           // tensor_dim1 hi=0 | tile_dim0 = 7056
        g1[4] = 1;                                    // tile_dim1 = 1 | tile_dim2 = 0 (unused)
        g1[5] = HW_;                                  // tensor_dim0_stride lo32
        g1[6] = (int)((uint32_t)HW_ << 16);           // stride0 hi=0 | tensor_dim1_stride lo
        g1[7] = 0;                                    // tensor_dim1_stride hi

        // D# groups 2/3: higher dims unused (tile_dim2..4 = 0)
        i32x4 g2; g2[0] = 1; g2[1] = 0; g2[2] = 0; g2[3] = 0;  // tensor_dim2 = 1
        i32x4 g3; g3[0] = 0; g3[1] = 0; g3[2] = 0; g3[3] = 0;

        // Extra group required by the 6-arg toolchain variant (unused: zero).
        i32x8 g4;
        g4[0] = 0; g4[1] = 0; g4[2] = 0; g4[3] = 0;
        g4[4] = 0; g4[5] = 0; g4[6] = 0; g4[7] = 0;

        __builtin_amdgcn_tensor_load_to_lds(g0, g1, g2, g3, g4, 0);
        __builtin_amdgcn_s_wait_tensorcnt(0);         // TENSORcnt == 0
    }
    __syncthreads();

    // ---- Reference fp32 arithmetic (matches _random_shift_aug) ----
    const float eps   = 1.0f / (float)HP_;                       // 1/92
    const float stepl = (2.0f - 2.0f * eps) / (float)(HP_ - 1);  // linspace step
    const float scale = 2.0f / (float)HP_;
    const float sxf = (float)sx * scale;
    const float syf = (float)sy * scale;

    float* const outplane = out + (size_t)(n * C_ + c) * HW_;

    for (int u = threadIdx.x; u < NU_; u += 256) {
        const int j  = u / W4_;           // output row
        const int i0 = (u - j * W4_) * 4; // first of 4 output columns

        // ---- y terms (shared by the 4 columns) ----
        const float ay = (-1.0f + eps) + (float)j * stepl;
        const float gy = ay + syf;
        const float iy = ((gy + 1.0f) * (float)HP_ - 1.0f) * 0.5f;
        const float iy0f = floorf(iy);
        const float wy = iy - iy0f;
        const int iy0 = (int)iy0f;
        const int cy0 = min(max(iy0     - PAD_, 0), H_ - 1);
        const int cy1 = min(max(iy0 + 1 - PAD_, 0), H_ - 1);
        const float vy0 = (iy0     >= 0 && iy0     < HP_) ? 1.0f : 0.0f;
        const float vy1 = (iy0 + 1 >= 0 && iy0 + 1 < HP_) ? 1.0f : 0.0f;
        const float* rowA = &smem[cy0 * W_];
        const float* rowB = &smem[cy1 * W_];

        float4 res;
        float* resf = (float*)&res;

#pragma unroll
        for (int k = 0; k < 4; ++k) {
            const int i = i0 + k;
            const float ax = (-1.0f + eps) + (float)i * stepl;
            const float gx = ax + sxf;
            const float ix = ((gx + 1.0f) * (float)HP_ - 1.0f) * 0.5f;
            const float ix0f = floorf(ix);
            const float wx = ix - ix0f;
            const int ix0 = (int)ix0f;
            const int cx0 = min(max(ix0     - PAD_, 0), W_ - 1);
            const int cx1 = min(max(ix0 + 1 - PAD_, 0), W_ - 1);
            const float vx0 = (ix0     >= 0 && ix0     < HP_) ? 1.0f : 0.0f;
            const float vx1 = (ix0 + 1 >= 0 && ix0 + 1 < HP_) ? 1.0f : 0.0f;

            const float v00 = rowA[cx0] * (vy0 * vx0);
            const float v01 = rowA[cx1] * (vy0 * vx1);
            const float v10 = rowB[cx0] * (vy1 * vx0);
            const float v11 = rowB[cx1] * (vy1 * vx1);

            resf[k] = v00 * (1.0f - wy) * (1.0f - wx)
                    + v01 * (1.0f - wy) * wx
                    + v10 * wy * (1.0f - wx)
                    + v11 * wy * wx;
        }

        *(float4*)(outplane + j * W_ + i0) = res;
    }
}

extern "C" void kernel_launch(void* const* d_in, const int* in_sizes, int n_in,
                              void* d_out, int out_size, void* d_ws, size_t ws_size,
                              hipStream_t stream) {
    const float* x     = (const float*)d_in[0];
    const int*   shift = (const int*)d_in[1];
    float*       out   = (float*)d_out;

    const int N = in_sizes[0] / (C_ * HW_);   // 1024
    dim3 grid(C_, N);
    rsa_tdm_kernel<<<grid, 256, 0, stream>>>(x, shift, out);
}